// InterpBaselineEncoder_83829171683905
// MI455X (gfx1250) — compile-verified
//
#include <hip/hip_runtime.h>
#include <stdint.h>

typedef __attribute__((ext_vector_type(2))) float v2f;
typedef __attribute__((ext_vector_type(8))) float v8f;

#define BB 4
#define HH 128
#define WW 128
#define UU 8192
#define TT 4096
#define YY 8
#define SS 1024      // coarse cells per batch (32x32)
#define SW 32
#define FH 4
#define FW 4

// ---------------------------------------------------------------------------
// K1: coarsen xc_on/yc_on by 4x4 mean pooling; zero the scatter accumulators.
// One thread per (b, cell).
// ---------------------------------------------------------------------------
__global__ void ibe_coarsen_zero(const float* __restrict__ xc_on,
                                 const float* __restrict__ yc_on,
                                 float* __restrict__ grid_loc,  // [B,S,2]
                                 float* __restrict__ grid_val,  // [B,S,Y]
                                 float* __restrict__ sums,      // [B,S,Y]
                                 float* __restrict__ counts) {  // [B,S]
  int tid = blockIdx.x * blockDim.x + threadIdx.x;
  if (tid >= BB * SS) return;
  int b = tid / SS, s = tid % SS;
  int sy = s / SW, sx = s % SW;
  float a0 = 0.f, a1 = 0.f, av[YY];
#pragma unroll
  for (int y = 0; y < YY; ++y) av[y] = 0.f;
  for (int i = 0; i < FH; ++i)
    for (int j = 0; j < FW; ++j) {
      int h = sy * FH + i, w = sx * FW + j;
      long base = ((long)(b * HH + h) * WW + w);
      a0 += xc_on[base * 2 + 0];
      a1 += xc_on[base * 2 + 1];
#pragma unroll
      for (int y = 0; y < YY; ++y) av[y] += yc_on[base * YY + y];
    }
  const float inv = 1.0f / (float)(FH * FW);
  grid_loc[(long)tid * 2 + 0] = a0 * inv;
  grid_loc[(long)tid * 2 + 1] = a1 * inv;
#pragma unroll
  for (int y = 0; y < YY; ++y) {
    grid_val[(long)tid * YY + y] = av[y] * inv;
    sums[(long)tid * YY + y] = 0.f;
  }
  counts[tid] = 0.f;
}

// ---------------------------------------------------------------------------
// K2: WMMA brute-force nearest-cell search.
// score[m][n] = -2*(p0*g0 + p1*g1) + (g0^2+g1^2)  ==  ||p-g||^2 - ||p||^2,
// so argmin_n(score) == argmin_n L2 == argmin_n L1 on this separable grid.
// A (16x4):  row m = [p0_m, p1_m, 1, 0]
//   (lanes 0-15 hold K=0,1; lanes 16-31 hold K=2,3 per the f32 A layout)
// B (4x16):  col n = [-2*g0_n, -2*g1_n, g0_n^2+g1_n^2, 0]
//   staged in LDS as sB[half][cell] = float2 so the per-iteration fetch is a
//   single non-divergent ds_load_b64 with a loop-invariant per-lane base.
// One wave per 16-point tile; 64 cell-tile iterations. blockDim must be 128.
// ---------------------------------------------------------------------------
__global__ __launch_bounds__(128) void ibe_nn_wmma(
    const float* __restrict__ pts,       // [B, N, 2]
    const float* __restrict__ grid_loc,  // [B, S, 2]
    int* __restrict__ nearest,           // [B, N]
    int nPerBatch) {
  __shared__ v2f sB[2 * SS];  // [0][n] = (-2g0, -2g1), [1][n] = (|g|^2, 0)

  const int tilesPerBlock = 4;
  int blocksPerBatch = nPerBatch / (16 * tilesPerBlock);
  int b = blockIdx.x / blocksPerBatch;
  int tile0 = (blockIdx.x % blocksPerBatch) * tilesPerBlock;

  for (int i = threadIdx.x; i < SS; i += blockDim.x) {
    float g0 = grid_loc[((long)b * SS + i) * 2 + 0];
    float g1 = grid_loc[((long)b * SS + i) * 2 + 1];
    v2f lo, hi2;
    lo.x = -2.0f * g0;
    lo.y = -2.0f * g1;
    hi2.x = g0 * g0 + g1 * g1;
    hi2.y = 0.0f;
    sB[i] = lo;
    sB[SS + i] = hi2;
  }
  __syncthreads();

  int wave = threadIdx.x >> 5;
  int lane = threadIdx.x & 31;
  int hi = lane >> 4;   // which 16-lane half
  int l16 = lane & 15;
  int tile = tile0 + wave;
  long pbase = (long)b * nPerBatch + (long)tile * 16;

  // A matrix: lanes 0-15 hold K=0,1 (coords of point m=l16); lanes 16-31 hold
  // K=2,3 (constants 1, 0).
  v2f a;
  if (hi == 0) {
    a.x = pts[(pbase + l16) * 2 + 0];
    a.y = pts[(pbase + l16) * 2 + 1];
  } else {
    a.x = 1.0f;
    a.y = 0.0f;
  }

  // Loop-invariant per-lane LDS base: cell n = nb + l16, row-pair = hi.
  const v2f* bptr = &sB[hi * SS + l16];

  float bestS[8];
  int bestN[8];
#pragma unroll
  for (int r = 0; r < 8; ++r) { bestS[r] = 3.0e38f; bestN[r] = 0; }

#pragma unroll 4
  for (int nb = 0; nb < SS; nb += 16) {
    int n = nb + l16;        // cell index this lane scores (both halves)
    v2f bm = bptr[nb];       // single ds_load_b64, no divergence
    v8f c = {0.f, 0.f, 0.f, 0.f, 0.f, 0.f, 0.f, 0.f};
    v8f d = __builtin_amdgcn_wmma_f32_16x16x4_f32(
        false, a, false, bm, (short)0, c, false, false);
    // d[r] = score[M = r + 8*hi][N = l16]; N fixed per lane -> scalar argmin.
#pragma unroll
    for (int r = 0; r < 8; ++r) {
      float sc = d[r];
      if (sc < bestS[r]) { bestS[r] = sc; bestN[r] = n; }
    }
  }

  // Butterfly argmin across the 16 lanes that hold the same point row.
  // xor masks 8,4,2,1 keep lanes within their 16-lane half (wave32).
#pragma unroll
  for (int r = 0; r < 8; ++r) {
    float sc = bestS[r];
    int idx = bestN[r];
#pragma unroll
    for (int m = 8; m >= 1; m >>= 1) {
      float os = __shfl_xor(sc, m, 32);
      int oi = __shfl_xor(idx, m, 32);
      if (os < sc || (os == sc && oi < idx)) { sc = os; idx = oi; }
    }
    if (l16 == 0) {
      int mrow = r + 8 * hi;  // lane 0 -> rows 0..7, lane 16 -> rows 8..15
      nearest[pbase + mrow] = idx;
    }
  }
}

// ---------------------------------------------------------------------------
// K3: scatter-add off-grid values into their nearest cells (float atomics).
// One thread per (b, u, y).
// ---------------------------------------------------------------------------
__global__ void ibe_scatter(const float* __restrict__ yc_off,  // [B,U,Y]
                            const int* __restrict__ nearest_u, // [B,U]
                            float* __restrict__ sums,          // [B,S,Y]
                            float* __restrict__ counts) {      // [B,S]
  long tid = (long)blockIdx.x * blockDim.x + threadIdx.x;
  if (tid >= (long)BB * UU * YY) return;
  int b = (int)(tid / ((long)UU * YY));
  long rem = tid % ((long)UU * YY);
  int u = (int)(rem / YY);
  int y = (int)(rem % YY);
  int s = nearest_u[(long)b * UU + u];
  atomicAdd(&sums[((long)b * SS + s) * YY + y], yc_off[tid]);
  if (y == 0) atomicAdd(&counts[(long)b * SS + s], 1.0f);
}

// ---------------------------------------------------------------------------
// K4: fused average + gather at target points. One thread per (b, t, y).
// ---------------------------------------------------------------------------
__global__ void ibe_gather(const float* __restrict__ sums,
                           const float* __restrict__ grid_val,
                           const float* __restrict__ counts,
                           const int* __restrict__ nearest_t,  // [B,T]
                           float* __restrict__ out) {          // [B,T,Y]
  long tid = (long)blockIdx.x * blockDim.x + threadIdx.x;
  if (tid >= (long)BB * TT * YY) return;
  int b = (int)(tid / ((long)TT * YY));
  long rem = tid % ((long)TT * YY);
  int t = (int)(rem / YY);
  int y = (int)(rem % YY);
  int s = nearest_t[(long)b * TT + t];
  long cs = ((long)b * SS + s) * YY + y;
  out[tid] = (sums[cs] + grid_val[cs]) / (counts[(long)b * SS + s] + 1.0f);
}

extern "C" void kernel_launch(void* const* d_in, const int* in_sizes, int n_in,
                              void* d_out, int out_size, void* d_ws, size_t ws_size,
                              hipStream_t stream) {
  const float* xc_off = (const float*)d_in[0];  // [B,U,2]
  const float* yc_off = (const float*)d_in[1];  // [B,U,Y]
  const float* xc_on  = (const float*)d_in[2];  // [B,H,W,2]
  const float* yc_on  = (const float*)d_in[3];  // [B,H,W,Y]
  const float* xt     = (const float*)d_in[4];  // [B,T,2]
  float* out = (float*)d_out;

  char* ws = (char*)d_ws;
  float* grid_loc = (float*)(ws + 0);                     //  32768 B
  float* grid_val = (float*)(ws + 32768);                 // 131072 B
  float* sums     = (float*)(ws + 163840);                // 131072 B
  float* counts   = (float*)(ws + 294912);                //  16384 B
  int* nearest_u  = (int*)(ws + 311296);                  // 131072 B
  int* nearest_t  = (int*)(ws + 442368);                  //  65536 B
  (void)ws_size; (void)in_sizes; (void)n_in; (void)out_size;

  // K1: coarsen + zero accumulators
  ibe_coarsen_zero<<<(BB * SS + 255) / 256, 256, 0, stream>>>(
      xc_on, yc_on, grid_loc, grid_val, sums, counts);

  // K2: WMMA nearest-cell search for off-grid points and targets
  ibe_nn_wmma<<<BB * (UU / 64), 128, 0, stream>>>(xc_off, grid_loc, nearest_u, UU);
  ibe_nn_wmma<<<BB * (TT / 64), 128, 0, stream>>>(xt, grid_loc, nearest_t, TT);

  // K3: scatter-mean accumulation
  ibe_scatter<<<(int)(((long)BB * UU * YY + 255) / 256), 256, 0, stream>>>(
      yc_off, nearest_u, sums, counts);

  // K4: fused average + gather
  ibe_gather<<<(int)(((long)BB * TT * YY + 255) / 256), 256, 0, stream>>>(
      sums, grid_val, counts, nearest_t, out);
}